// GATModel_18777597018172
// MI455X (gfx1250) — compile-verified
//
#include <hip/hip_runtime.h>
#include <math.h>

typedef __attribute__((ext_vector_type(2))) float v2f;
typedef __attribute__((ext_vector_type(8))) float v8f;

#define KN 50000          // nodes
#define KE 800000         // edges (before self loops)
#define KET (KE + KN)     // edges incl. self loops
#define KG 64             // graphs
#define KC 64             // channels per head
#define KSLOPE 0.2f
#define KEPS 1e-16f

// ---------- order-preserving float <-> uint for atomic max ----------
__device__ __forceinline__ unsigned f2key(float f) {
    unsigned b = __float_as_uint(f);
    return (b & 0x80000000u) ? ~b : (b | 0x80000000u);
}
__device__ __forceinline__ float key2f(unsigned k) {
    unsigned b = (k & 0x80000000u) ? (k ^ 0x80000000u) : ~k;
    return __uint_as_float(b);
}

// ---------- generic zero fill ----------
__global__ void zero_kernel(float* __restrict__ p, long n) {
    long t = (long)blockIdx.x * blockDim.x + threadIdx.x;
    if (t < n) p[t] = 0.0f;
}

// ---------- WMMA f32 GEMM: C[M,N] = A[M,K] * B[K,N] ----------
// wave32: each wave computes a 16x64 tile using V_WMMA_F32_16X16X4_F32.
// A fragment (16x4): lanes 0-15 hold K={0,1}, lanes 16-31 hold K={2,3}.
// B fragment (4x16): lane n holds B[k0][n], B[k0+1][n] with k0 = 2*(lane>>4).
// C/D (16x16): VGPR r -> row r (lanes 0-15) / row r+8 (lanes 16-31).
__global__ void gemm_wmma_kernel(const float* __restrict__ A,
                                 const float* __restrict__ B,
                                 float* __restrict__ C,
                                 int M, int K, int N) {
    const int lane = threadIdx.x;            // 0..31
    const int wave = threadIdx.y;            // 0..7
    const int row0 = (blockIdx.x * 8 + wave) * 16;
    if (row0 >= M) return;                   // wave-uniform exit (EXEC stays all-1s for WMMA)
    const int col0 = blockIdx.y * 64;

    const int m  = lane & 15;
    const int kh = (lane >> 4) * 2;

    v8f acc0 = {}, acc1 = {}, acc2 = {}, acc3 = {};
    const float* arow = A + (size_t)(row0 + m) * K + kh;

    for (int k = 0; k < K; k += 4) {
        v2f a = *(const v2f*)(arow + k);     // 8B aligned: (k+kh) is even
        const float* bp = B + (size_t)(k + kh) * N + col0 + m;
        v2f b0, b1, b2, b3;
        b0.x = bp[0];  b0.y = bp[N];
        b1.x = bp[16]; b1.y = bp[N + 16];
        b2.x = bp[32]; b2.y = bp[N + 32];
        b3.x = bp[48]; b3.y = bp[N + 48];
        acc0 = __builtin_amdgcn_wmma_f32_16x16x4_f32(false, a, false, b0, (short)0, acc0, false, false);
        acc1 = __builtin_amdgcn_wmma_f32_16x16x4_f32(false, a, false, b1, (short)0, acc1, false, false);
        acc2 = __builtin_amdgcn_wmma_f32_16x16x4_f32(false, a, false, b2, (short)0, acc2, false, false);
        acc3 = __builtin_amdgcn_wmma_f32_16x16x4_f32(false, a, false, b3, (short)0, acc3, false, false);
    }

    const int rbase = row0 + ((lane >> 4) * 8);
#pragma unroll
    for (int r = 0; r < 8; ++r) {
        float* crow = C + (size_t)(rbase + r) * N + col0 + m;
        crow[0]  = acc0[r];
        crow[16] = acc1[r];
        crow[32] = acc2[r];
        crow[48] = acc3[r];
    }
}

// ---------- per-node attention logits: als/ald[n,h] = sum_c h[n,h,c]*a[h,c] ----------
__global__ void attn_logits_kernel(const float* __restrict__ h,
                                   const float* __restrict__ a_src,
                                   const float* __restrict__ a_dst,
                                   float* __restrict__ als,
                                   float* __restrict__ ald,
                                   int heads) {
    int t = blockIdx.x * blockDim.x + threadIdx.x;
    int total = KN * heads;
    if (t >= total) return;
    int n = t / heads, hd = t - n * heads;
    const float* hp = h + (size_t)n * heads * KC + (size_t)hd * KC;
    const float* as = a_src + hd * KC;
    const float* ad = a_dst + hd * KC;
    float s1 = 0.f, s2 = 0.f;
#pragma unroll 8
    for (int c = 0; c < KC; ++c) { float v = hp[c]; s1 += v * as[c]; s2 += v * ad[c]; }
    als[t] = s1;
    ald[t] = s2;
}

// ---------- edge pass 1: leaky-relu logits + segment max over dst ----------
__global__ void edge_logit_max_kernel(const int* __restrict__ ei,
                                      const float* __restrict__ als,
                                      const float* __restrict__ ald,
                                      float* __restrict__ el,
                                      unsigned* __restrict__ mkey,
                                      int heads) {
    int t = blockIdx.x * blockDim.x + threadIdx.x;
    int total = KET * heads;
    if (t >= total) return;
    int e = t / heads, hd = t - e * heads;
    int s, d;
    if (e < KE) { s = ei[e]; d = ei[KE + e]; } else { s = e - KE; d = s; }
    float x = als[s * heads + hd] + ald[d * heads + hd];
    x = (x > 0.f) ? x : KSLOPE * x;
    el[t] = x;
    atomicMax(&mkey[d * heads + hd], f2key(x));
}

// ---------- edge pass 2: p = exp(x - max); segment sum over dst ----------
__global__ void edge_exp_sum_kernel(const int* __restrict__ ei,
                                    float* __restrict__ el,
                                    const unsigned* __restrict__ mkey,
                                    float* __restrict__ ssum,
                                    int heads) {
    int t = blockIdx.x * blockDim.x + threadIdx.x;
    int total = KET * heads;
    if (t >= total) return;
    int e = t / heads, hd = t - e * heads;
    int d = (e < KE) ? ei[KE + e] : (e - KE);
    float p = expf(el[t] - key2f(mkey[d * heads + hd]));
    el[t] = p;
    atomicAdd(&ssum[d * heads + hd], p);
}

// ---------- edge pass 3: agg[dst] += alpha * h[src] (one wave per edge) ----------
__global__ void edge_aggregate_kernel(const int* __restrict__ ei,
                                      const float* __restrict__ h,
                                      const float* __restrict__ el,
                                      const float* __restrict__ ssum,
                                      float* __restrict__ agg,
                                      int heads) {
    int gw = (blockIdx.x * blockDim.x + threadIdx.x) >> 5;   // global wave id
    int lane = threadIdx.x & 31;
    if (gw >= KET) return;                                   // wave-uniform
    int e = gw;
    int s, d;
    if (e < KE) { s = ei[e]; d = ei[KE + e]; } else { s = e - KE; d = s; }
    const int HC = heads * KC;
    const float* hs = h + (size_t)s * HC;
    float* ag = agg + (size_t)d * HC;
    for (int f = 0; f < HC / 32; ++f) {
        int col = f * 32 + lane;
        int hd = col >> 6;                                   // KC == 64
        float alpha = el[e * heads + hd] / (ssum[d * heads + hd] + KEPS);
        atomicAdd(&ag[col], alpha * hs[col]);
    }
}

// ---------- bias + optional relu ----------
__global__ void bias_act_kernel(const float* __restrict__ agg,
                                const float* __restrict__ bias,
                                float* __restrict__ out,
                                int HC, int do_relu) {
    long t = (long)blockIdx.x * blockDim.x + threadIdx.x;
    long total = (long)KN * HC;
    if (t >= total) return;
    float v = agg[t] + bias[t % HC];
    if (do_relu) v = v > 0.f ? v : 0.f;
    out[t] = v;
}

// ---------- global mean pool: sums + counts ----------
__global__ void pool_kernel(const float* __restrict__ h,
                            const int* __restrict__ batch,
                            float* __restrict__ gsum,
                            float* __restrict__ gcnt) {
    int t = blockIdx.x * blockDim.x + threadIdx.x;
    if (t >= KN * KC) return;
    int n = t >> 6, c = t & 63;
    int g = batch[n];
    atomicAdd(&gsum[g * KC + c], h[t]);
    if (c == 0) atomicAdd(&gcnt[g], 1.0f);
}

// ---------- final tiny linear: out = (gsum/cnt) @ lin_W + lin_b ----------
__global__ void final_linear_kernel(const float* __restrict__ gsum,
                                    const float* __restrict__ gcnt,
                                    const float* __restrict__ W,
                                    const float* __restrict__ b,
                                    float* __restrict__ out) {
    int t = blockIdx.x * blockDim.x + threadIdx.x;
    if (t >= KG * KC) return;
    int g = t >> 6, o = t & 63;
    float cnt = gcnt[g]; cnt = cnt > 1.f ? cnt : 1.f;
    float inv = 1.f / cnt;
    float acc = 0.f;
#pragma unroll 8
    for (int k = 0; k < KC; ++k) acc += (gsum[g * KC + k] * inv) * W[k * KC + o];
    out[t] = acc + b[o];
}

// ================= host-side layer driver =================
static void run_gat_layer(const float* xin, int K,
                          const float* W, const float* asrc, const float* adst,
                          const float* bias, int heads, int do_relu,
                          float* hbuf, float* aggbuf, float* el, unsigned* mkey,
                          float* ssum, float* als, float* ald,
                          const int* ei, float* xout, hipStream_t stream) {
    const int HC = heads * KC;
    // 1) h = x @ W  (WMMA)
    {
        dim3 blk(32, 8);
        dim3 grd((KN / 16 + 7) / 8, HC / 64);
        gemm_wmma_kernel<<<grd, blk, 0, stream>>>(xin, W, hbuf, KN, K, HC);
    }
    // 2) per-node logits
    attn_logits_kernel<<<(KN * heads + 255) / 256, 256, 0, stream>>>(hbuf, asrc, adst, als, ald, heads);
    // 3) clear accumulators
    zero_kernel<<<((long)KN * HC + 255) / 256, 256, 0, stream>>>(aggbuf, (long)KN * HC);
    zero_kernel<<<(KN * heads + 255) / 256, 256, 0, stream>>>((float*)mkey, KN * heads); // bits 0 = minimal key
    zero_kernel<<<(KN * heads + 255) / 256, 256, 0, stream>>>(ssum, KN * heads);
    // 4) edge softmax + aggregate
    edge_logit_max_kernel<<<(KET * heads + 255) / 256, 256, 0, stream>>>(ei, als, ald, el, mkey, heads);
    edge_exp_sum_kernel<<<(KET * heads + 255) / 256, 256, 0, stream>>>(ei, el, mkey, ssum, heads);
    edge_aggregate_kernel<<<((long)KET * 32 + 255) / 256, 256, 0, stream>>>(ei, hbuf, el, ssum, aggbuf, heads);
    // 5) bias + activation -> next layer input
    bias_act_kernel<<<((long)KN * HC + 255) / 256, 256, 0, stream>>>(aggbuf, bias, xout, HC, do_relu);
}

extern "C" void kernel_launch(void* const* d_in, const int* in_sizes, int n_in,
                              void* d_out, int out_size, void* d_ws, size_t ws_size,
                              hipStream_t stream) {
    (void)in_sizes; (void)n_in; (void)out_size; (void)ws_size;

    const float* x      = (const float*)d_in[0];    // [N,128]
    const int*   ei     = (const int*)d_in[1];      // [2,E]
    /* d_in[2] edge_attr unused */
    const int*   batch  = (const int*)d_in[3];      // [N]
    const float* W1     = (const float*)d_in[4];    // [128,256]
    const float* a1s    = (const float*)d_in[5];
    const float* a1d    = (const float*)d_in[6];
    const float* b1     = (const float*)d_in[7];
    const float* W2     = (const float*)d_in[8];    // [256,256]
    const float* a2s    = (const float*)d_in[9];
    const float* a2d    = (const float*)d_in[10];
    const float* b2     = (const float*)d_in[11];
    const float* W3     = (const float*)d_in[12];   // [256,64]
    const float* a3s    = (const float*)d_in[13];
    const float* a3d    = (const float*)d_in[14];
    const float* b3     = (const float*)d_in[15];
    const float* linW   = (const float*)d_in[16];   // [64,64]
    const float* linb   = (const float*)d_in[17];
    float* out = (float*)d_out;                     // [64,64]

    // ---- workspace carve (~170.5 MB) ----
    float* cur  = (float*)d_ws;                     // N*256  (layer io; also reused for layer3 [N,64] output)
    float* hbuf = cur  + (size_t)KN * 256;          // N*256
    float* agg  = hbuf + (size_t)KN * 256;          // N*256
    float* el   = agg  + (size_t)KN * 256;          // ET*4
    unsigned* mkey = (unsigned*)(el + (size_t)KET * 4); // N*4
    float* ssum = (float*)(mkey + (size_t)KN * 4);  // N*4
    float* als  = ssum + (size_t)KN * 4;            // N*4
    float* ald  = als  + (size_t)KN * 4;            // N*4
    float* gsum = ald  + (size_t)KN * 4;            // 64*64
    float* gcnt = gsum + KG * KC;                   // 64

    // layer 1: in x[N,128] -> cur[N,256]
    run_gat_layer(x,   128, W1, a1s, a1d, b1, 4, 1, hbuf, agg, el, mkey, ssum, als, ald, ei, cur, stream);
    // layer 2: cur[N,256] -> cur[N,256]
    run_gat_layer(cur, 256, W2, a2s, a2d, b2, 4, 1, hbuf, agg, el, mkey, ssum, als, ald, ei, cur, stream);
    // layer 3: cur[N,256] -> cur[N,64] (heads=1, no relu)
    run_gat_layer(cur, 256, W3, a3s, a3d, b3, 1, 0, hbuf, agg, el, mkey, ssum, als, ald, ei, cur, stream);

    // global mean pool + final linear
    zero_kernel<<<(KG * KC + KG + 255) / 256, 256, 0, stream>>>(gsum, KG * KC + KG); // gsum + gcnt contiguous
    pool_kernel<<<(KN * KC + 255) / 256, 256, 0, stream>>>(cur, batch, gsum, gcnt);
    final_linear_kernel<<<(KG * KC + 255) / 256, 256, 0, stream>>>(gsum, gcnt, linW, linb, out);
}